// Projector_37726992728176
// MI455X (gfx1250) — compile-verified
//
#include <hip/hip_runtime.h>

typedef float v2f __attribute__((ext_vector_type(2)));
typedef float v8f __attribute__((ext_vector_type(8)));

#define SIDE 128
#define TILE 16

// One wave (32 lanes) per block. Each wave owns a 16x16 output tile of one
// batch image. WMMA f32 16x16x4 computes the per-pixel ray base coordinates;
// the D-matrix layout (lane l, vgpr r -> pixel m=r+8*(l>=16), n=l%16) defines
// which 8 rays each lane marches.
__global__ __launch_bounds__(32) void projector_raymarch(
    const float* __restrict__ rotmat,   // [B,3,3] row-major
    const float* __restrict__ vol,      // [128,128,128] (D,H,W)
    float* __restrict__ out)            // [B,1,128,128]
{
    const int lane = threadIdx.x;          // 0..31
    const int blk  = blockIdx.x;
    const int b    = blk >> 6;             // batch
    const int tile = blk & 63;
    const int ti   = (tile >> 3) * TILE;   // tile origin, output row i
    const int tj   = (tile & 7)  * TILE;   // tile origin, output col j

    const float* __restrict__ R = rotmat + b * 9;
    const float R00 = R[0], R01 = R[1], R02 = R[2];
    const float R10 = R[3], R11 = R[4], R12 = R[5];
    const float R20 = R[6], R21 = R[7], R22 = R[8];

    const float HALF  = 63.5f;
    const float LSTEP = 2.0f / 127.0f;

    const int  n  = lane & 15;
    const int  hi = lane >> 4;             // 0 for lanes 0-15, 1 for 16-31
    const bool lo = (hi == 0);

    // lin coordinates for this tile
    const float lin_i = -1.0f + LSTEP * (float)(ti + lane);  // A rows (lanes 0-15)
    const float lin_j = -1.0f + LSTEP * (float)(tj + n);     // B cols

    // base_c(i,j) = lin_i * (63.5*R1c)  +  1 * (63.5*(lin_j*R0c + 1 - R2c))
    // Expressed as D = A(16x4) x B(4x16) with K=0,1 active, K=2,3 zero.
    v8f dX, dY, dZ;
#if __has_builtin(__builtin_amdgcn_wmma_f32_16x16x4_f32)
    v2f a;                                  // A 16x4: lanes0-15 V0=K0,V1=K1
    a.x = lo ? lin_i : 0.0f;
    a.y = lo ? 1.0f  : 0.0f;

    v2f bx, by, bz;                         // B 4x16: lanes0-15 V0=K0,V1=K1 (N=lane)
    bx.x = lo ? HALF * R10 : 0.0f;
    bx.y = lo ? HALF * (lin_j * R00 + 1.0f - R20) : 0.0f;
    by.x = lo ? HALF * R11 : 0.0f;
    by.y = lo ? HALF * (lin_j * R01 + 1.0f - R21) : 0.0f;
    bz.x = lo ? HALF * R12 : 0.0f;
    bz.y = lo ? HALF * (lin_j * R02 + 1.0f - R22) : 0.0f;

    v8f cz = {};
    dX = __builtin_amdgcn_wmma_f32_16x16x4_f32(false, a, false, bx, (short)0, cz, false, false);
    dY = __builtin_amdgcn_wmma_f32_16x16x4_f32(false, a, false, by, (short)0, cz, false, false);
    dZ = __builtin_amdgcn_wmma_f32_16x16x4_f32(false, a, false, bz, (short)0, cz, false, false);
#else
    // VALU fallback with identical (m,n)->lane/vgpr mapping
    #pragma unroll
    for (int r = 0; r < 8; ++r) {
        const float li = -1.0f + LSTEP * (float)(ti + r + 8 * hi);
        dX[r] = li * (HALF * R10) + HALF * (lin_j * R00 + 1.0f - R20);
        dY[r] = li * (HALF * R11) + HALF * (lin_j * R01 + 1.0f - R21);
        dZ[r] = li * (HALF * R12) + HALF * (lin_j * R02 + 1.0f - R22);
    }
#endif

    // step per k in unnormalized voxel space is exactly R row 2
    const float sx = R20, sy = R21, sz = R22;

    float px[8], py[8], pz[8], acc[8];
    #pragma unroll
    for (int r = 0; r < 8; ++r) {
        px[r] = dX[r]; py[r] = dY[r]; pz[r] = dZ[r]; acc[r] = 0.0f;
    }

    #pragma unroll 1
    for (int k = 0; k < SIDE; ++k) {
        #pragma unroll
        for (int r = 0; r < 8; ++r) {
            const float x = px[r], y = py[r], z = pz[r];
            px[r] = x + sx; py[r] = y + sy; pz[r] = z + sz;

            // any-tap-valid bounding test (coherent across the tile -> execz skip)
            if (x > -1.0f && x < 128.0f &&
                y > -1.0f && y < 128.0f &&
                z > -1.0f && z < 128.0f) {
                const float fx = floorf(x), fy = floorf(y), fz = floorf(z);
                const float tx = x - fx,    ty = y - fy,    tz = z - fz;
                const int   x0 = (int)fx,   y0 = (int)fy,   z0 = (int)fz;

                // validity folded into the 1-D weights (zeros padding)
                const float wx0 = (x0 >= 0)  ? (1.0f - tx) : 0.0f;
                const float wx1 = (x0 < 127) ? tx          : 0.0f;
                const float wy0 = (y0 >= 0)  ? (1.0f - ty) : 0.0f;
                const float wy1 = (y0 < 127) ? ty          : 0.0f;
                const float wz0 = (z0 >= 0)  ? (1.0f - tz) : 0.0f;
                const float wz1 = (z0 < 127) ? tz          : 0.0f;

                const int xc0 = (x0 < 0) ? 0 : x0;
                const int xc1 = (x0 >= 127) ? 127 : x0 + 1;
                const int yc0 = (y0 < 0) ? 0 : y0;
                const int yc1 = (y0 >= 127) ? 127 : y0 + 1;
                const int zc0 = (z0 < 0) ? 0 : z0;
                const int zc1 = (z0 >= 127) ? 127 : z0 + 1;

                const int b00 = ((zc0 << 7) + yc0) << 7;
                const int b01 = ((zc0 << 7) + yc1) << 7;
                const int b10 = ((zc1 << 7) + yc0) << 7;
                const int b11 = ((zc1 << 7) + yc1) << 7;

                const float v000 = vol[b00 + xc0], v001 = vol[b00 + xc1];
                const float v010 = vol[b01 + xc0], v011 = vol[b01 + xc1];
                const float v100 = vol[b10 + xc0], v101 = vol[b10 + xc1];
                const float v110 = vol[b11 + xc0], v111 = vol[b11 + xc1];

                const float r00 = fmaf(v001, wx1, v000 * wx0);
                const float r01 = fmaf(v011, wx1, v010 * wx0);
                const float r10 = fmaf(v101, wx1, v100 * wx0);
                const float r11 = fmaf(v111, wx1, v110 * wx0);
                const float p0  = fmaf(r01, wy1, r00 * wy0);
                const float p1  = fmaf(r11, wy1, r10 * wy0);
                acc[r] = fmaf(p1, wz1, fmaf(p0, wz0, acc[r]));
            }
        }
    }

    // D layout: vgpr r, lanes0-15 -> M=r ; lanes16-31 -> M=r+8 ; N=lane%16
    float* __restrict__ o = out + b * (SIDE * SIDE) + (tj + n);
    const int i0 = ti + 8 * hi;
    #pragma unroll
    for (int r = 0; r < 8; ++r) {
        o[(i0 + r) * SIDE] = acc[r];
    }
}

extern "C" void kernel_launch(void* const* d_in, const int* in_sizes, int n_in,
                              void* d_out, int out_size, void* d_ws, size_t ws_size,
                              hipStream_t stream) {
    (void)n_in; (void)out_size; (void)d_ws; (void)ws_size;
    const float* rotmat = (const float*)d_in[0];   // [B,3,3] fp32
    const float* vol    = (const float*)d_in[1];   // [128,128,128] fp32
    float*       out    = (float*)d_out;           // [B,1,128,128] fp32

    const int B = in_sizes[0] / 9;                 // 16
    const int tiles_per_img = (SIDE / TILE) * (SIDE / TILE);  // 64
    dim3 grid(B * tiles_per_img);                  // 1024 one-wave blocks
    dim3 block(32);
    projector_raymarch<<<grid, block, 0, stream>>>(rotmat, vol, out);
}